// GatedSpatialMoE2d_7971459301717
// MI455X (gfx1250) — compile-verified
//
#include <hip/hip_runtime.h>

typedef float v2f __attribute__((ext_vector_type(2)));
typedef float v8f __attribute__((ext_vector_type(8)));

#define HW    16384   // 128*128
#define CIN   64
#define NE    16
#define NC    64
#define TOPK  4
#define NPIX  131072  // B(8) * H(128) * W(128)

// ---------------------------------------------------------------------------
// Kernel 1: gate GEMM via V_WMMA_F32_16X16X4_F32, then softmax + top-4 routing.
// One wave computes a 16-pixel x 16-expert logit tile (K=Cin=64 in 16 steps).
// Block = 256 threads = 8 waves = 128 pixels.
// ---------------------------------------------------------------------------
__global__ __launch_bounds__(256) void moe_gate_route(
    const float* __restrict__ x,       // (B, Cin, H, W)
    const float* __restrict__ gate_w,  // (E, Cin)
    const float* __restrict__ gate_b,  // (E,)
    float4* __restrict__ rt_w,         // (NPIX,) packed top-4 weights
    int4*   __restrict__ rt_i)         // (NPIX,) packed top-4 indices
{
    __shared__ float slog[128 * 17];   // [pixel-in-block][expert], pad 17

    const int t    = threadIdx.x;
    const int wave = t >> 5;
    const int lane = t & 31;
    const int hf   = lane >> 4;        // lane half (0: lanes 0-15, 1: 16-31)
    const int li   = lane & 15;

    // This wave's 16-pixel group (all pixels share the same batch b:
    // HW is a multiple of the 128-pixel block tile).
    const int g0  = blockIdx.x * 128 + wave * 16;  // global pixel id
    const int b   = g0 / HW;
    const int hw0 = g0 % HW;
    const float* xb = x + (size_t)b * CIN * HW + hw0 + li;  // M = li

    v8f acc = {0.f, 0.f, 0.f, 0.f, 0.f, 0.f, 0.f, 0.f};
    #pragma unroll
    for (int k0 = 0; k0 < CIN; k0 += 4) {
        // ISA layout, 32-bit A 16x4: v0 holds K=k0 (lanes 0-15) / K=k0+2
        // (lanes 16-31); v1 holds K=k0+1 / K=k0+3.  B (4x16) mirrors it.
        const int c = k0 + 2 * hf;
        v2f a, bf;
        a.x  = xb[(size_t)c * HW];          // x[b, c,   hw0+li]
        a.y  = xb[(size_t)(c + 1) * HW];    // x[b, c+1, hw0+li]
        bf.x = gate_w[li * CIN + c];        // gate_w[e=li, c]   (4KB, cache-hot)
        bf.y = gate_w[li * CIN + c + 1];
        acc = __builtin_amdgcn_wmma_f32_16x16x4_f32(
            /*neg_a=*/false, a, /*neg_b=*/false, bf,
            /*c_mod=*/(short)0, acc, /*reuse_a=*/false, /*reuse_b=*/false);
    }

    // D layout: VGPR r -> (M = r + 8*hf, N = li). Scatter logits to LDS.
    #pragma unroll
    for (int r = 0; r < 8; ++r)
        slog[(wave * 16 + r + 8 * hf) * 17 + li] = acc[r];
    __syncthreads();

    // Per-pixel softmax over 16 experts + top-4 (ties -> lowest index,
    // matching lax.top_k). Threads 0..127 each own one pixel.
    if (t < 128) {
        float l[NE];
        float mx = -3.402823466e38f;
        #pragma unroll
        for (int e = 0; e < NE; ++e) {
            l[e] = slog[t * 17 + e] + gate_b[e];
            mx = fmaxf(mx, l[e]);
        }
        float s = 0.f;
        #pragma unroll
        for (int e = 0; e < NE; ++e) { l[e] = __expf(l[e] - mx); s += l[e]; }
        const float inv = 1.f / s;

        float w[TOPK]; int id[TOPK];
        #pragma unroll
        for (int j = 0; j < TOPK; ++j) {
            float best = -1.f; int bi = 0;
            #pragma unroll
            for (int e = 0; e < NE; ++e)
                if (l[e] > best) { best = l[e]; bi = e; }
            w[j] = best * inv; id[j] = bi; l[bi] = -1.f;  // exp() >= 0 > -1
        }
        const int pix = blockIdx.x * 128 + t;
        rt_w[pix] = make_float4(w[0], w[1], w[2], w[3]);
        rt_i[pix] = make_int4(id[0], id[1], id[2], id[3]);
    }
}

// ---------------------------------------------------------------------------
// Kernel 2: gather the 4 selected expert rows per pixel (contiguous 256B,
// coalesced float4 reads), weighted-sum in registers, transpose through LDS,
// write channels-first output with pixel-contiguous (coalesced) stores.
// Block = 256 threads handles a 64-pixel x 64-channel tile.
// ---------------------------------------------------------------------------
__global__ __launch_bounds__(256) void moe_gather_sum(
    const float*  __restrict__ experts, // (B, E, H, W, C)
    const float4* __restrict__ rt_w,
    const int4*   __restrict__ rt_i,
    float* __restrict__ out)            // (B, C, H, W)
{
    __shared__ float sw[64][4];
    __shared__ int   si[64][4];
    __shared__ float tile[64 * 65];     // [pixel][channel], pad 65

    const int t    = threadIdx.x;
    const int pix0 = blockIdx.x * 64;
    const int b    = pix0 / HW;
    const int hw0  = pix0 % HW;

    if (t < 64) {
        float4 w4 = rt_w[pix0 + t];
        int4   i4 = rt_i[pix0 + t];
        sw[t][0] = w4.x; sw[t][1] = w4.y; sw[t][2] = w4.z; sw[t][3] = w4.w;
        si[t][0] = i4.x; si[t][1] = i4.y; si[t][2] = i4.z; si[t][3] = i4.w;
    }
    __syncthreads();

    // 16 lanes (cg) cover one 64-channel expert row as float4s -> 256B
    // contiguous per (pixel, expert): fully coalesced HBM reads.
    const int cg = t & 15;
    const int pg = t >> 4;
    #pragma unroll
    for (int pp = 0; pp < 4; ++pp) {
        const int p = pp * 16 + pg;
        float ax = 0.f, ay = 0.f, az = 0.f, aw = 0.f;
        #pragma unroll
        for (int j = 0; j < TOPK; ++j) {
            const float wgt = sw[p][j];
            const int   e   = si[p][j];
            const float4 v = *(const float4*)(experts +
                ((size_t)(b * NE + e) * HW + (size_t)(hw0 + p)) * NC + cg * 4);
            ax += wgt * v.x; ay += wgt * v.y; az += wgt * v.z; aw += wgt * v.w;
        }
        float* tr = &tile[p * 65 + cg * 4];
        tr[0] = ax; tr[1] = ay; tr[2] = az; tr[3] = aw;
    }
    __syncthreads();

    // Writeback: consecutive threads -> consecutive pixels for a fixed
    // channel => coalesced 256B stores into (B, C, H, W).
    const int p  = t & 63;
    const int cb = t >> 6;
    #pragma unroll
    for (int i = 0; i < 16; ++i) {
        const int c = cb * 16 + i;
        out[(size_t)(b * NC + c) * HW + (size_t)(hw0 + p)] = tile[p * 65 + c];
    }
}

// ---------------------------------------------------------------------------
extern "C" void kernel_launch(void* const* d_in, const int* in_sizes, int n_in,
                              void* d_out, int out_size, void* d_ws, size_t ws_size,
                              hipStream_t stream) {
    const float* x       = (const float*)d_in[0];  // (8, 64, 128, 128)
    const float* gate_w  = (const float*)d_in[1];  // (16, 64)
    const float* gate_b  = (const float*)d_in[2];  // (16,)
    const float* experts = (const float*)d_in[3];  // (8, 16, 128, 128, 64)
    // d_in[4] is k == 4 (compile-time TOPK)

    // Workspace: 131072 * (float4 + int4) = 4 MB of routing data.
    float4* rt_w = (float4*)d_ws;
    int4*   rt_i = (int4*)((char*)d_ws + (size_t)NPIX * sizeof(float4));

    moe_gate_route<<<NPIX / 128, 256, 0, stream>>>(x, gate_w, gate_b, rt_w, rt_i);
    moe_gather_sum<<<NPIX / 64, 256, 0, stream>>>(experts, rt_w, rt_i, (float*)d_out);
}